// AVWGCN_88338887344445
// MI455X (gfx1250) — compile-verified
//
#include <hip/hip_runtime.h>

// MI455X / gfx1250, wave32. f32 WMMA path: V_WMMA_F32_16X16X4_F32.
typedef float v2f __attribute__((ext_vector_type(2)));
typedef float v8f __attribute__((ext_vector_type(8)));

#define B_    64
#define N_    8192
#define DIN_  64
#define DOUT_ 64
#define D_    10
#define C_    4
#define NPC_  2048

#define ZS   68    // Z tile row stride (floats), padded: bank = (4*row + col) % 64 -> conflict-free A loads
#define WS   68    // W tile row stride
#define SBS  650   // sum_EH per-b stride in LDS: bank = (10*b + i) % 64 -> conflict-free across b
#define NODES_PER_WG 16
#define GRP  4     // nodes whose W tiles are built per weights_pool sweep (4x L2 reuse)

// ---------- kernel 0: zero sum_EH scratch ----------
__global__ void k_zero(float* __restrict__ p, int n) {
  int i = blockIdx.x * blockDim.x + threadIdx.x;
  if (i < n) p[i] = 0.f;
}

// ---------- kernel 1: sum_EH[b][d][i] = sum_p relu(E[g(p),d]) * x[b,g(p),i] * mask[b,g(p)] ----------
// grid: (B, N/256), block 256. Memory-bound pass over x (134 MB). f32 atomics into [64][10][64].
__global__ __launch_bounds__(256) void k_sumEH(const float* __restrict__ x,
                                               const float* __restrict__ emb,
                                               const float* __restrict__ mask,
                                               const int*   __restrict__ nodes,
                                               float* __restrict__ gsum) {
  const int b  = blockIdx.x;
  const int t  = threadIdx.x;
  const int i  = t & 63;
  const int dg = t >> 6;                 // uniform per wave (wave32)
  const int d0 = dg * 3;                 // dg=0..2 -> 3 d's, dg=3 -> d=9 only
  float a0 = 0.f, a1 = 0.f, a2 = 0.f;
  const int p0 = blockIdx.y * 256;
  for (int pp = 0; pp < 256; ++pp) {
    const int g = nodes[p0 + pp];
    const float m  = mask[(size_t)b * N_ + g];
    const float xv = x[((size_t)b * N_ + g) * DIN_ + i] * m;
    const float* e = emb + (size_t)g * D_;
    a0 += fmaxf(e[d0], 0.f) * xv;
    if (dg < 3) {
      a1 += fmaxf(e[d0 + 1], 0.f) * xv;
      a2 += fmaxf(e[d0 + 2], 0.f) * xv;
    }
  }
  atomicAdd(&gsum[(b * D_ + d0) * DIN_ + i], a0);
  if (dg < 3) {
    atomicAdd(&gsum[(b * D_ + d0 + 1) * DIN_ + i], a1);
    atomicAdd(&gsum[(b * D_ + d0 + 2) * DIN_ + i], a2);
  }
}

// ---------- kernel 2: per-node W build + Z build + (Z @ W + bb) via v_wmma_f32_16x16x4_f32 ----------
// grid: (C, NPC/16), block 256 (8 waves). LDS ~249 KB: sum_EH resident, W tiles built for 4 nodes
// per weights_pool sweep, Z tile per node, 16 WMMA output tiles split 2 per wave.
__global__ __launch_bounds__(256) void k_main(const float* __restrict__ x,
                                              const float* __restrict__ emb,
                                              const float* __restrict__ mask,
                                              const float* __restrict__ wpool,
                                              const float* __restrict__ bpool,
                                              const int*   __restrict__ nodes,
                                              const float* __restrict__ gsum,
                                              float* __restrict__ out) {
  __shared__ float sEH[B_ * SBS];          // 166400 B  sum_EH[b][d][i], stride-650 per b
  __shared__ float Zt[B_ * ZS];            //  17408 B  Z[b][i]
  __shared__ float Wt[GRP * DIN_ * WS];    //  69632 B  W[jj][i][o]
  __shared__ float sbb[GRP * DOUT_];       //   1024 B  bias per node
  __shared__ float sE[GRP * D_];           // raw E (for W, bias)
  __shared__ float stE[GRP * D_];          // relu(E) (for Z)
  __shared__ int   sg[GRP];

  const int t     = threadIdx.x;
  const int c     = blockIdx.x;            // client
  const int base  = c * NPC_ + blockIdx.y * NODES_PER_WG;
  const int lane  = t & 31;
  const int w     = t >> 5;                // wave id 0..7

  // one-time: sum_EH -> LDS (160 KB, reused by all 16 nodes of this WG)
  for (int k = t; k < B_ * D_ * DIN_; k += 256) {
    int b = k / (D_ * DIN_);
    int r = k - b * (D_ * DIN_);
    sEH[b * SBS + r] = gsum[k];
  }
  __syncthreads();

  for (int j0 = 0; j0 < NODES_PER_WG; j0 += GRP) {
    // --- embeddings for GRP nodes ---
    if (t < GRP * D_) {
      int jj = t / D_, d = t - jj * D_;
      int g = nodes[base + j0 + jj];
      if (d == 0) sg[jj] = g;
      float e = emb[(size_t)g * D_ + d];
      sE[jj * D_ + d]  = e;                // raw E for W / bias
      stE[jj * D_ + d] = fmaxf(e, 0.f);    // relu(E) for Z
    }
    __syncthreads();

    // --- bias: bb[jj][o] = sum_d E*bpool[c,d,o] ---
    {
      int jj = t >> 6, o = t & 63;
      float acc = 0.f;
      #pragma unroll
      for (int d = 0; d < D_; ++d)
        acc += sE[jj * D_ + d] * bpool[((size_t)c * D_ + d) * DOUT_ + o];
      sbb[jj * DOUT_ + o] = acc;
    }
    // --- W tiles for 4 nodes in one sweep of weights_pool (each L2 element used 4x) ---
    for (int io = t; io < DIN_ * DOUT_; io += 256) {
      int i = io >> 6, o = io & 63;
      float a0 = 0.f, a1 = 0.f, a2 = 0.f, a3 = 0.f;
      #pragma unroll
      for (int d = 0; d < D_; ++d) {
        float s = wpool[(((size_t)c * D_ + d) * DIN_ + i) * DOUT_ + o];
        a0 += sE[0 * D_ + d] * s;
        a1 += sE[1 * D_ + d] * s;
        a2 += sE[2 * D_ + d] * s;
        a3 += sE[3 * D_ + d] * s;
      }
      Wt[0 * DIN_ * WS + i * WS + o] = a0;
      Wt[1 * DIN_ * WS + i * WS + o] = a1;
      Wt[2 * DIN_ * WS + i * WS + o] = a2;
      Wt[3 * DIN_ * WS + i * WS + o] = a3;
    }
    __syncthreads();

    for (int jj = 0; jj < GRP; ++jj) {
      const int g = sg[jj];
      const int p = base + j0 + jj;        // output node position
      // --- Z tile: Z[b][i] = x*mask + sum_d relu(E)*sum_EH[b][d][i] ---
      {
        const int b  = t >> 2;
        const int ib = (t & 3) << 4;
        float te[D_];
        #pragma unroll
        for (int d = 0; d < D_; ++d) te[d] = stE[jj * D_ + d];
        const float m = mask[(size_t)b * N_ + g];
        const float* xb = x + ((size_t)b * N_ + g) * DIN_;
        #pragma unroll 4
        for (int k = 0; k < 16; ++k) {
          const int i = ib + k;
          float z = xb[i] * m;
          const float* se = &sEH[b * SBS + i];
          #pragma unroll
          for (int d = 0; d < D_; ++d) z += te[d] * se[d * DIN_];
          Zt[b * ZS + i] = z;
        }
      }
      __syncthreads();

      // --- 64x64x64 GEMM: 16 tiles of 16x16, 2 per wave, K via 16x wmma_f32_16x16x4 ---
      const int mb    = w & 3;
      const int nb0   = (w >> 2) * 2;
      const int arow  = mb * 16 + (lane & 15);
      const int khalf = (lane >> 4) * 2;                 // A/B K sub-offset per lane half
      const int bcol0 = nb0 * 16 + (lane & 15);
      const int bcol1 = bcol0 + 16;
      v8f acc0 = {}; v8f acc1 = {};
      #pragma unroll
      for (int kb = 0; kb < 16; ++kb) {
        const int k0 = kb * 4 + khalf;
        v2f a = *(const v2f*)&Zt[arow * ZS + k0];        // A: Z[M=lane%16][K pair]
        const float* wb = &Wt[jj * DIN_ * WS + k0 * WS]; // B: W[K rows][N=lane%16]
        v2f b0; b0.x = wb[bcol0]; b0.y = wb[WS + bcol0];
        v2f b1; b1.x = wb[bcol1]; b1.y = wb[WS + bcol1];
        acc0 = __builtin_amdgcn_wmma_f32_16x16x4_f32(false, a, false, b0, (short)0, acc0, false, false);
        acc1 = __builtin_amdgcn_wmma_f32_16x16x4_f32(false, a, false, b1, (short)0, acc1, false, false);
      }
      // --- store with bias (C/D layout: VGPR r -> M=r (+8 for hi lanes), N=lane%16) ---
      const float bias0 = sbb[jj * DOUT_ + bcol0];
      const float bias1 = sbb[jj * DOUT_ + bcol1];
      const int brow = mb * 16 + (lane >> 4) * 8;
      #pragma unroll
      for (int r = 0; r < 8; ++r) {
        size_t off = ((size_t)(brow + r) * N_ + p) * DOUT_;
        out[off + bcol0] = acc0[r] + bias0;
        out[off + bcol1] = acc1[r] + bias1;
      }
      __syncthreads();   // protect Zt (and, on last jj, sE/Wt) before overwrite
    }
  }
}

extern "C" void kernel_launch(void* const* d_in, const int* in_sizes, int n_in,
                              void* d_out, int out_size, void* d_ws, size_t ws_size,
                              hipStream_t stream) {
  const float* x     = (const float*)d_in[0];
  const float* emb   = (const float*)d_in[1];
  // d_in[2] = poly_coefficients (unused in sprtrelu mode)
  const float* mask  = (const float*)d_in[3];
  const float* wpool = (const float*)d_in[4];
  const float* bpool = (const float*)d_in[5];
  const int*   nodes = (const int*)d_in[6];
  float* out  = (float*)d_out;
  float* gsum = (float*)d_ws;                       // [B][D][DIN] f32 = 160 KB scratch

  const int nsum = B_ * D_ * DIN_;
  k_zero <<<(nsum + 255) / 256, 256, 0, stream>>>(gsum, nsum);
  k_sumEH<<<dim3(B_, N_ / 256), 256, 0, stream>>>(x, emb, mask, nodes, gsum);
  k_main <<<dim3(C_, NPC_ / NODES_PER_WG), 256, 0, stream>>>(x, emb, mask, wpool, bpool,
                                                             nodes, gsum, out);
}